// Critic_MATD3_Graph_16767552323670
// MI455X (gfx1250) — compile-verified
//
#include <hip/hip_runtime.h>

// MI455X / gfx1250, wave32. fp32 network -> V_WMMA_F32_16X16X4_F32 for all GEMMs.
// GEMMs: 512-thread blocks (16 waves), 32 rows x 256 cols per block.
// A panel staged into LDS with GLOBAL_LOAD_ASYNC_TO_LDS_B128 (ASYNCcnt-tracked,
// no VGPR round trip), then read back with conflict-free ds_load_b64 pairs.

typedef __attribute__((ext_vector_type(2))) float v2f;
typedef __attribute__((ext_vector_type(8))) float v8f;

#define OBS 128
#define ACT 64
#define INDIM 192
#define HID 256
#define NEG_SLOPE 0.2f

#define LDSK256 260   // 256 + 4: bank = (4*mr + ka) % 64, distinct per mr
#define LDSK192 196   // 192 + 4: same property (192 % 64 == 0)
// Row strides in bytes: 260*4=1040, 196*4=784 -> both multiples of 16 (B128-safe)

// ---- monotone float <-> uint key for atomicMax-based segment max ----
__device__ __forceinline__ unsigned f2key(float f) {
  unsigned u = __float_as_uint(f);
  return (u & 0x80000000u) ? ~u : (u | 0x80000000u);
}
__device__ __forceinline__ float key2f(unsigned k) {
  unsigned u = (k & 0x80000000u) ? (k ^ 0x80000000u) : ~k;
  return __uint_as_float(u);
}
__device__ __forceinline__ float leaky(float v) {
  return v > 0.f ? v : NEG_SLOPE * v;
}
__device__ __forceinline__ v8f wmma_f32(v2f a, v2f b, v8f c) {
  return __builtin_amdgcn_wmma_f32_16x16x4_f32(
      /*neg_a=*/false, a, /*neg_b=*/false, b,
      /*c_mod=*/(short)0, c, /*reuse_a=*/false, /*reuse_b=*/false);
}

// Async 16-byte copy global -> LDS (gfx1250 GLOBAL_LOAD_ASYNC_TO_LDS_B128,
// GV addressing mode: per-lane 64-bit global addr, per-lane 32-bit LDS addr).
__device__ __forceinline__ void async_g2l_b128(void* lds_dst, const void* gsrc) {
  const unsigned laddr = (unsigned)(size_t)lds_dst;   // generic ptr low 32b == LDS addr
  asm volatile("global_load_async_to_lds_b128 %0, %1, off"
               :: "v"(laddr), "v"(gsrc)
               : "memory");
}
__device__ __forceinline__ void async_wait_all() {
  asm volatile("s_wait_asynccnt 0x0" ::: "memory");
}

// ============================================================================
// GEMM 1: X = [s | a] @ W_gat   ([N,192] x [192,256] -> [N,256])
// Block: 32 rows x 256 cols, 16 waves. Requires N % 32 == 0 (100000/32 = 3125).
// ============================================================================
__global__ __launch_bounds__(512) void gemm_gat_kernel(
    const float* __restrict__ S, const float* __restrict__ A,
    const float* __restrict__ W, float* __restrict__ X, int N) {
  __shared__ float As[32][LDSK192];          // ~25 KB of the 320 KB WGP LDS

  const int lane = threadIdx.x;              // 0..31
  const int wid  = threadIdx.y;              // 0..15 -> col tile
  const int half = lane >> 4;
  const int mr   = lane & 15;
  const int rowBase = blockIdx.x * 32;
  const int colBase = wid * 16;

  // ---- async stage of the concatenated [s|a] panel (memory -> LDS direct) ----
  {
    const int t = wid * 32 + lane;           // 0..511
    const int r = t >> 4;                    // 0..31 (16 threads per row)
    // s part: 32 x 128 floats, 8 per thread (2x B128)
    {
      const int c0 = (t & 15) * 8;
      const float* gsrc = S + (size_t)(rowBase + r) * OBS + c0;
      async_g2l_b128(&As[r][c0],     gsrc);
      async_g2l_b128(&As[r][c0 + 4], gsrc + 4);
    }
    // a part: 32 x 64 floats, 4 per thread (1x B128)
    {
      const int c0 = (t & 15) * 4;
      async_g2l_b128(&As[r][OBS + c0], A + (size_t)(rowBase + r) * ACT + c0);
    }
  }
  // prefetch this wave's weight slice while the async engine stages A
  for (int kk = 0; kk < INDIM; kk += 32)
    __builtin_prefetch(W + (size_t)kk * HID + colBase + mr, 0, 3);
  async_wait_all();
  __syncthreads();

  v8f c0 = {}, c1 = {};
  for (int k0 = 0; k0 < INDIM; k0 += 4) {
    const int ka = k0 + half * 2;
    const float2 t0 = *(const float2*)&As[mr][ka];        // row tile 0
    const float2 t1 = *(const float2*)&As[16 + mr][ka];   // row tile 1
    v2f a0, a1, bv;
    a0.x = t0.x; a0.y = t0.y;
    a1.x = t1.x; a1.y = t1.y;
    bv.x = W[(size_t)ka * HID + colBase + mr];
    bv.y = W[(size_t)(ka + 1) * HID + colBase + mr];
    c0 = wmma_f32(a0, bv, c0);
    c1 = wmma_f32(a1, bv, c1);
  }
#pragma unroll
  for (int r = 0; r < 8; ++r) {
    const int rr = r + half * 8;
    X[(size_t)(rowBase + rr) * HID + colBase + mr]      = c0[r];
    X[(size_t)(rowBase + 16 + rr) * HID + colBase + mr] = c1[r];
  }
}

// ============================================================================
// GEMM 2: Y = act(Xin @ W + bias)   ([N,256] x [256,256])
// ============================================================================
__global__ __launch_bounds__(512) void gemm256_kernel(
    const float* __restrict__ Xin, const float* __restrict__ W,
    const float* __restrict__ bias, float* __restrict__ Y, int N, int do_relu) {
  __shared__ float As[32][LDSK256];          // ~33 KB

  const int lane = threadIdx.x;
  const int wid  = threadIdx.y;
  const int half = lane >> 4;
  const int mr   = lane & 15;
  const int rowBase = blockIdx.x * 32;
  const int colBase = wid * 16;

  // ---- async stage: 32 rows x 256 cols = 8192 floats, 4x B128 per thread ----
  {
    const int t = wid * 32 + lane;
    const int r = t >> 4;
    const int c0 = (t & 15) * 16;
    const float* gsrc = Xin + (size_t)(rowBase + r) * HID + c0;
#pragma unroll
    for (int q = 0; q < 4; ++q)
      async_g2l_b128(&As[r][c0 + q * 4], gsrc + q * 4);
  }
  for (int kk = 0; kk < HID; kk += 32)
    __builtin_prefetch(W + (size_t)kk * HID + colBase + mr, 0, 3);
  async_wait_all();
  __syncthreads();

  v8f c0 = {}, c1 = {};
  for (int k0 = 0; k0 < HID; k0 += 4) {
    const int ka = k0 + half * 2;
    const float2 t0 = *(const float2*)&As[mr][ka];
    const float2 t1 = *(const float2*)&As[16 + mr][ka];
    v2f a0, a1, bv;
    a0.x = t0.x; a0.y = t0.y;
    a1.x = t1.x; a1.y = t1.y;
    bv.x = W[(size_t)ka * HID + colBase + mr];
    bv.y = W[(size_t)(ka + 1) * HID + colBase + mr];
    c0 = wmma_f32(a0, bv, c0);
    c1 = wmma_f32(a1, bv, c1);
  }
  const float bval = bias[colBase + mr];
#pragma unroll
  for (int r = 0; r < 8; ++r) {
    const int rr = r + half * 8;
    float v0 = c0[r] + bval;
    float v1 = c1[r] + bval;
    if (do_relu) { v0 = fmaxf(v0, 0.f); v1 = fmaxf(v1, 0.f); }
    Y[(size_t)(rowBase + rr) * HID + colBase + mr]      = v0;
    Y[(size_t)(rowBase + 16 + rr) * HID + colBase + mr] = v1;
  }
}

// ============================================================================
// Per-row attention dots: a_s[row] = x[row]·att_src, a_d[row] = x[row]·att_dst
// ============================================================================
__global__ __launch_bounds__(256) void att_kernel(
    const float* __restrict__ X, const float* __restrict__ wsrc,
    const float* __restrict__ wdst, float* __restrict__ a_s,
    float* __restrict__ a_d, int N) {
  const int lane = threadIdx.x & 31;
  const int row  = blockIdx.x * (blockDim.x >> 5) + (threadIdx.x >> 5);
  if (row >= N) return;
  const float* xr = X + (size_t)row * HID;
  float ss = 0.f, sd = 0.f;
#pragma unroll
  for (int j = 0; j < 8; ++j) {
    const int cidx = lane + j * 32;
    const float v = xr[cidx];
    ss += v * wsrc[cidx];
    sd += v * wdst[cidx];
  }
#pragma unroll
  for (int off = 16; off > 0; off >>= 1) {
    ss += __shfl_xor(ss, off, 32);
    sd += __shfl_xor(sd, off, 32);
  }
  if (lane == 0) { a_s[row] = ss; a_d[row] = sd; }
}

// ============================================================================
// Init: h = broadcast(bias_gat); mkey = 0 (== -inf sentinel); denom = 0
// ============================================================================
__global__ void init_kernel(float* __restrict__ H, const float* __restrict__ bias,
                            unsigned* __restrict__ mkey, float* __restrict__ denom,
                            int N) {
  const size_t i = (size_t)blockIdx.x * blockDim.x + threadIdx.x;
  const size_t tot = (size_t)N * HID;
  if (i < tot) H[i] = bias[i & (HID - 1)];
  if (i < (size_t)N) { mkey[i] = 0u; denom[i] = 0.f; }
}

// ============================================================================
// Edge pass 1: segment max of leaky(a_s[src]+a_d[dst]) into mkey[dst]
// ============================================================================
__global__ void edge_max_kernel(const int* __restrict__ src, const int* __restrict__ dst,
                                const float* __restrict__ a_s, const float* __restrict__ a_d,
                                unsigned* __restrict__ mkey, int E) {
  const int e = blockIdx.x * blockDim.x + threadIdx.x;
  if (e >= E) return;
  const int s = src[e], d = dst[e];
  const float v = leaky(a_s[s] + a_d[d]);
  atomicMax(&mkey[d], f2key(v));
}

__global__ void key2m_kernel(const unsigned* __restrict__ mkey,
                             float* __restrict__ m, int N) {
  const int i = blockIdx.x * blockDim.x + threadIdx.x;
  if (i >= N) return;
  const unsigned k = mkey[i];
  m[i] = (k == 0u) ? 0.f : key2f(k);   // sentinel == no in-edges -> m = 0
}

// ============================================================================
// Edge pass 2: denom[dst] += exp(e - m[dst])
// ============================================================================
__global__ void edge_denom_kernel(const int* __restrict__ src, const int* __restrict__ dst,
                                  const float* __restrict__ a_s, const float* __restrict__ a_d,
                                  const float* __restrict__ m, float* __restrict__ denom,
                                  int E) {
  const int e = blockIdx.x * blockDim.x + threadIdx.x;
  if (e >= E) return;
  const int s = src[e], d = dst[e];
  const float v = leaky(a_s[s] + a_d[d]);
  atomicAdd(&denom[d], __expf(v - m[d]));
}

// ============================================================================
// Edge pass 3: h[dst] += alpha * x[src]  (wave per edge, 8 channels per lane)
// x (102 MB) is L2-resident on MI455X (192 MB L2): gathers + atomics hit L2.
// ============================================================================
__global__ __launch_bounds__(256) void edge_scatter_kernel(
    const int* __restrict__ src, const int* __restrict__ dst,
    const float* __restrict__ a_s, const float* __restrict__ a_d,
    const float* __restrict__ m, const float* __restrict__ denom,
    const float* __restrict__ X, float* __restrict__ H, int E) {
  const int e = (int)(((size_t)blockIdx.x * blockDim.x + threadIdx.x) >> 5);
  const int lane = threadIdx.x & 31;
  if (e >= E) return;
  const int s = src[e], d = dst[e];
  const float v = leaky(a_s[s] + a_d[d]);
  const float alpha = __expf(v - m[d]) / (denom[d] + 1e-16f);
  const float* xs = X + (size_t)s * HID;
  float* hd = H + (size_t)d * HID;
#pragma unroll
  for (int j = 0; j < 8; ++j) {
    const int cidx = lane + j * 32;
    atomicAdd(&hd[cidx], alpha * xs[cidx]);
  }
}

// ============================================================================
// Final layer: out[row] = h[row]·W3 + b3   (wave per row)
// ============================================================================
__global__ __launch_bounds__(256) void rowdot_kernel(
    const float* __restrict__ H, const float* __restrict__ W3,
    const float* __restrict__ b3, float* __restrict__ out, int N) {
  const int lane = threadIdx.x & 31;
  const int row  = blockIdx.x * (blockDim.x >> 5) + (threadIdx.x >> 5);
  if (row >= N) return;
  const float* hr = H + (size_t)row * HID;
  float sum = 0.f;
#pragma unroll
  for (int j = 0; j < 8; ++j) {
    const int cidx = lane + j * 32;
    sum += hr[cidx] * W3[cidx];
  }
#pragma unroll
  for (int off = 16; off > 0; off >>= 1) sum += __shfl_xor(sum, off, 32);
  if (lane == 0) out[row] = sum + b3[0];
}

// ============================================================================
extern "C" void kernel_launch(void* const* d_in, const int* in_sizes, int n_in,
                              void* d_out, int out_size, void* d_ws, size_t ws_size,
                              hipStream_t stream) {
  const float* S    = (const float*)d_in[0];
  const float* A    = (const float*)d_in[1];
  const int*   EI   = (const int*)d_in[2];
  const float* Wg   = (const float*)d_in[3];
  const float* atts = (const float*)d_in[4];
  const float* attd = (const float*)d_in[5];
  const float* bg   = (const float*)d_in[6];
  const float* q1W1 = (const float*)d_in[7],  *q1b1 = (const float*)d_in[8];
  const float* q1W2 = (const float*)d_in[9],  *q1b2 = (const float*)d_in[10];
  const float* q1W3 = (const float*)d_in[11], *q1b3 = (const float*)d_in[12];
  const float* q2W1 = (const float*)d_in[13], *q2b1 = (const float*)d_in[14];
  const float* q2W2 = (const float*)d_in[15], *q2b2 = (const float*)d_in[16];
  const float* q2W3 = (const float*)d_in[17], *q2b3 = (const float*)d_in[18];

  const int N = in_sizes[0] / OBS;     // 100000 (divisible by 32)
  const int E = in_sizes[2] / 2;       // 1600000
  const int* esrc = EI;
  const int* edst = EI + E;

  // Workspace layout (floats): X | H | T | a_s | a_d | m | mkey | denom
  float* ws = (float*)d_ws;
  const size_t NC = (size_t)N * HID;
  float*    X     = ws;                 // GAT features; reused as MLP layer-1 act
  float*    H     = ws + NC;            // GAT output (bias pre-seeded)
  float*    T     = ws + 2 * NC;        // MLP layer-2 act
  float*    a_s   = ws + 3 * NC;
  float*    a_d   = a_s + N;
  float*    m     = a_d + N;
  unsigned* mkey  = (unsigned*)(m + N);
  float*    denom = (float*)(mkey + N);
  float*    out   = (float*)d_out;      // [q1 (N) | q2 (N)]

  const dim3 gblk(32, 16, 1);                      // 16 waves: all 256 cols
  const dim3 ggrd((unsigned)(N / 32), 1, 1);       // 32 rows per block

  // 1) x = [s|a] @ W_gat
  gemm_gat_kernel<<<ggrd, gblk, 0, stream>>>(S, A, Wg, X, N);
  // 2) per-node attention logits
  att_kernel<<<(N + 7) / 8, 256, 0, stream>>>(X, atts, attd, a_s, a_d, N);
  // 3) h = bias; zero segment-max keys and denominators
  init_kernel<<<(unsigned)((NC + 255) / 256), 256, 0, stream>>>(H, bg, mkey, denom, N);
  // 4-7) segment softmax + weighted scatter
  edge_max_kernel<<<(E + 255) / 256, 256, 0, stream>>>(esrc, edst, a_s, a_d, mkey, E);
  key2m_kernel<<<(N + 255) / 256, 256, 0, stream>>>(mkey, m, N);
  edge_denom_kernel<<<(E + 255) / 256, 256, 0, stream>>>(esrc, edst, a_s, a_d, m, denom, E);
  edge_scatter_kernel<<<(E + 7) / 8, 256, 0, stream>>>(esrc, edst, a_s, a_d, m, denom, X, H, E);
  // 8-10) q1 MLP (X reused as scratch)
  gemm256_kernel<<<ggrd, gblk, 0, stream>>>(H, q1W1, q1b1, X, N, 1);
  gemm256_kernel<<<ggrd, gblk, 0, stream>>>(X, q1W2, q1b2, T, N, 1);
  rowdot_kernel<<<(N + 7) / 8, 256, 0, stream>>>(T, q1W3, q1b3, out, N);
  // 11-13) q2 MLP
  gemm256_kernel<<<ggrd, gblk, 0, stream>>>(H, q2W1, q2b1, X, N, 1);
  gemm256_kernel<<<ggrd, gblk, 0, stream>>>(X, q2W2, q2b2, T, N, 1);
  rowdot_kernel<<<(N + 7) / 8, 256, 0, stream>>>(T, q2W3, q2b3, out + N, N);
}